// Sampler_50706383897220
// MI455X (gfx1250) — compile-verified
//
#include <hip/hip_runtime.h>

typedef float f4 __attribute__((ext_vector_type(4)));

#define NEG_INF (-__builtin_inff())

constexpr int B       = 256;
constexpr int V       = 128000;
constexpr int THREADS = 256;            // 8 waves (wave32)
constexpr int WAVES   = THREADS / 32;
constexpr int F4_ROW  = V / 4;          // 32000 float4 per row
constexpr int F4_WAVE = F4_ROW / WAVES; // 4000 float4 per wave
constexpr int TILES   = F4_WAVE / 32;   // 125 tiles (32 lanes x float4 each)
constexpr int DEPTH   = 16;             // async pipeline depth per wave (8 KB in flight)

static_assert(TILES * 32 == F4_WAVE, "tiling must be exact");
static_assert((DEPTH & (DEPTH - 1)) == 0, "DEPTH power of two");
static_assert(TILES > DEPTH, "need steady state");

// ---- CDNA5 async global->LDS copy (ASYNCcnt path), GVS addressing mode ----
__device__ __forceinline__ void async_issue(unsigned lds_addr, unsigned voff,
                                            unsigned long long sbase) {
  asm volatile("global_load_async_to_lds_b128 %0, %1, %2"
               :: "v"(lds_addr), "v"(voff), "s"(sbase) : "memory");
}
// Steady-state reissue: tile t+DEPTH reuses the buffer whose ds_load was just
// issued for tile t, so retire wave's LDS reads before the async write can land.
__device__ __forceinline__ void async_issue_guarded(unsigned lds_addr, unsigned voff,
                                                    unsigned long long sbase) {
  asm volatile("s_wait_dscnt 0x0\n\t"
               "global_load_async_to_lds_b128 %0, %1, %2"
               :: "v"(lds_addr), "v"(voff), "s"(sbase) : "memory");
}

__device__ __forceinline__ void wait_async_full() {
  asm volatile("s_wait_asynccnt 0xf" ::: "memory");  // DEPTH-1
}
__device__ __forceinline__ void wait_async_le(int n) {  // drain epilogue only
  switch (n) {
    case 0:  asm volatile("s_wait_asynccnt 0x0" ::: "memory"); break;
    case 1:  asm volatile("s_wait_asynccnt 0x1" ::: "memory"); break;
    case 2:  asm volatile("s_wait_asynccnt 0x2" ::: "memory"); break;
    case 3:  asm volatile("s_wait_asynccnt 0x3" ::: "memory"); break;
    case 4:  asm volatile("s_wait_asynccnt 0x4" ::: "memory"); break;
    case 5:  asm volatile("s_wait_asynccnt 0x5" ::: "memory"); break;
    case 6:  asm volatile("s_wait_asynccnt 0x6" ::: "memory"); break;
    case 7:  asm volatile("s_wait_asynccnt 0x7" ::: "memory"); break;
    case 8:  asm volatile("s_wait_asynccnt 0x8" ::: "memory"); break;
    case 9:  asm volatile("s_wait_asynccnt 0x9" ::: "memory"); break;
    case 10: asm volatile("s_wait_asynccnt 0xa" ::: "memory"); break;
    case 11: asm volatile("s_wait_asynccnt 0xb" ::: "memory"); break;
    case 12: asm volatile("s_wait_asynccnt 0xc" ::: "memory"); break;
    case 13: asm volatile("s_wait_asynccnt 0xd" ::: "memory"); break;
    case 14: asm volatile("s_wait_asynccnt 0xe" ::: "memory"); break;
    default: asm volatile("s_wait_asynccnt 0xf" ::: "memory"); break;
  }
}

__global__ __launch_bounds__(THREADS)
void Sampler_50706383897220_kernel(const float* __restrict__ logits,
                                   const float* __restrict__ eps_cut,
                                   const float* __restrict__ eta_cut,
                                   const float* __restrict__ min_p,
                                   const float* __restrict__ gumbel,
                                   float* __restrict__ out_logits,
                                   float* __restrict__ out_tokens) {
  __shared__ f4 stage[WAVES * DEPTH * 32];   // 64 KB staging (of 320 KB LDS)
  __shared__ float red_f0[WAVES], red_f1[WAVES];
  __shared__ int   red_i[WAVES];
  __shared__ float bc_m0, bc_t0, bc_T;
  __shared__ int   bc_i0;

  const int row  = blockIdx.x;
  const int tid  = threadIdx.x;
  const int w    = tid >> 5;
  const int lane = tid & 31;

  const unsigned long long rowbase =
      (unsigned long long)(const void*)(logits + (size_t)row * V);
  const unsigned stage_base = (unsigned)(unsigned long long)(const void*)&stage[0];
  const unsigned my_lds = stage_base + (unsigned)((w * DEPTH * 32 + lane) * 16);
  const int my_f4 = w * F4_WAVE + lane;  // + t*32 per tile
  const int my_slot = w * DEPTH * 32 + lane;  // + buf*32

  // =========== Pass A: online (rowmax, Z0 = sum exp(x-m0), argmax) ==========
  // 4 independent accumulator chains (one per float4 lane) for TRANS-op ILP.
  float am[4] = {NEG_INF, NEG_INF, NEG_INF, NEG_INF};
  float as[4] = {0.0f, 0.0f, 0.0f, 0.0f};
  int   ai[4] = {0x7fffffff, 0x7fffffff, 0x7fffffff, 0x7fffffff};
  {
#pragma unroll
    for (int b = 0; b < DEPTH; ++b)
      async_issue(my_lds + b * 512u, (unsigned)((my_f4 + b * 32) * 16), rowbase);
    int t = 0;
    for (; t < TILES - DEPTH; ++t) {            // steady state
      wait_async_full();
      f4 v = stage[my_slot + (t & (DEPTH - 1)) * 32];
      int gi = (my_f4 + t * 32) * 4;
#pragma unroll
      for (int j = 0; j < 4; ++j) {
        float x = v[j];
        if (x > am[j]) { as[j] = as[j] * __expf(am[j] - x) + 1.0f; am[j] = x; ai[j] = gi + j; }
        else           { as[j] += __expf(x - am[j]); }
      }
      int tn = t + DEPTH;
      async_issue_guarded(my_lds + (unsigned)((tn & (DEPTH - 1)) * 512),
                          (unsigned)((my_f4 + tn * 32) * 16), rowbase);
    }
    for (; t < TILES; ++t) {                    // drain
      wait_async_le(TILES - t - 1);
      f4 v = stage[my_slot + (t & (DEPTH - 1)) * 32];
      int gi = (my_f4 + t * 32) * 4;
#pragma unroll
      for (int j = 0; j < 4; ++j) {
        float x = v[j];
        if (x > am[j]) { as[j] = as[j] * __expf(am[j] - x) + 1.0f; am[j] = x; ai[j] = gi + j; }
        else           { as[j] += __expf(x - am[j]); }
      }
    }
  }
  // merge the 4 lanes' accumulators
  float m = am[0], s = as[0]; int mi = ai[0];
#pragma unroll
  for (int j = 1; j < 4; ++j) {
    float M = fmaxf(m, am[j]);
    s  = s * __expf(m - M) + as[j] * __expf(am[j] - M);
    mi = (am[j] > m) ? ai[j] : ((m > am[j]) ? mi : (ai[j] < mi ? ai[j] : mi));
    m  = M;
  }
#pragma unroll
  for (int off = 16; off >= 1; off >>= 1) {
    float mo = __shfl_xor(m, off, 32);
    float so = __shfl_xor(s, off, 32);
    int   io = __shfl_xor(mi, off, 32);
    float M = fmaxf(m, mo);
    s  = s * __expf(m - M) + so * __expf(mo - M);
    mi = (mo > m) ? io : ((m > mo) ? mi : (io < mi ? io : mi));
    m  = M;
  }
  if (lane == 0) { red_f0[w] = m; red_f1[w] = s; red_i[w] = mi; }
  __syncthreads();
  if (tid == 0) {
    float M = red_f0[0], S = red_f1[0]; int I = red_i[0];
    for (int i = 1; i < WAVES; ++i) {
      float mo = red_f0[i], so = red_f1[i]; int io = red_i[i];
      float Mn = fmaxf(M, mo);
      S = S * __expf(M - Mn) + so * __expf(mo - Mn);
      I = (mo > M) ? io : ((M > mo) ? I : (io < I ? io : I));
      M = Mn;
    }
    bc_m0 = M; bc_i0 = I;
    bc_t0 = M + __logf(eps_cut[row] * S);   // eps==0 -> -inf -> keep all
  }
  __syncthreads();
  const float m0 = bc_m0, t0 = bc_t0;
  const int   i0 = bc_i0;

  // =========== Pass C: Z1 and S1 over epsilon survivors =====================
  float az[4] = {0.0f, 0.0f, 0.0f, 0.0f};
  float al[4] = {0.0f, 0.0f, 0.0f, 0.0f};
  {
#pragma unroll
    for (int b = 0; b < DEPTH; ++b)
      async_issue(my_lds + b * 512u, (unsigned)((my_f4 + b * 32) * 16), rowbase);
    int t = 0;
    for (; t < TILES - DEPTH; ++t) {
      wait_async_full();
      f4 v = stage[my_slot + (t & (DEPTH - 1)) * 32];
      int gi = (my_f4 + t * 32) * 4;
#pragma unroll
      for (int j = 0; j < 4; ++j) {
        float x = v[j];
        if ((x >= t0) || (gi + j == i0)) {
          float e = __expf(x - m0);
          az[j] += e;
          al[j] += e * (x - m0);
        }
      }
      int tn = t + DEPTH;
      async_issue_guarded(my_lds + (unsigned)((tn & (DEPTH - 1)) * 512),
                          (unsigned)((my_f4 + tn * 32) * 16), rowbase);
    }
    for (; t < TILES; ++t) {
      wait_async_le(TILES - t - 1);
      f4 v = stage[my_slot + (t & (DEPTH - 1)) * 32];
      int gi = (my_f4 + t * 32) * 4;
#pragma unroll
      for (int j = 0; j < 4; ++j) {
        float x = v[j];
        if ((x >= t0) || (gi + j == i0)) {
          float e = __expf(x - m0);
          az[j] += e;
          al[j] += e * (x - m0);
        }
      }
    }
  }
  float z1 = (az[0] + az[1]) + (az[2] + az[3]);
  float s1 = (al[0] + al[1]) + (al[2] + al[3]);
#pragma unroll
  for (int off = 16; off >= 1; off >>= 1) {
    z1 += __shfl_xor(z1, off, 32);
    s1 += __shfl_xor(s1, off, 32);
  }
  if (lane == 0) { red_f0[w] = z1; red_f1[w] = s1; }
  __syncthreads();
  if (tid == 0) {
    float Z1 = 0.0f, S1 = 0.0f;
    for (int i = 0; i < WAVES; ++i) { Z1 += red_f0[i]; S1 += red_f1[i]; }
    float neg_ent = S1 / Z1 - __logf(Z1);          // Z1 >= 1 (argmax term)
    float eta  = eta_cut[row];
    float eeta = fminf(eta, sqrtf(eta) * __expf(neg_ent));
    float t1 = m0 + __logf(eeta * Z1);             // eta==0 -> -inf
    float t2 = m0 + __logf(min_p[row]);            // min_p==0 -> -inf
    bc_T = fmaxf(t0, fmaxf(t1, t2));
  }
  __syncthreads();
  const float T = bc_T;

  // =========== Pass D: final mask, NT store, Gumbel-max argmax ==============
  const f4* grow = (const f4*)(gumbel + (size_t)row * V);
  f4*       orow = (f4*)(out_logits + (size_t)row * V);
  float abm[4] = {NEG_INF, NEG_INF, NEG_INF, NEG_INF};
  int   abi[4] = {0x7fffffff, 0x7fffffff, 0x7fffffff, 0x7fffffff};
  {
#pragma unroll
    for (int b = 0; b < DEPTH; ++b)
      async_issue(my_lds + b * 512u, (unsigned)((my_f4 + b * 32) * 16), rowbase);
    int t = 0;
    for (; t < TILES - DEPTH; ++t) {
      wait_async_full();
      f4 v = stage[my_slot + (t & (DEPTH - 1)) * 32];
      int f4i = my_f4 + t * 32;
      f4 g = __builtin_nontemporal_load(grow + f4i);  // read-once: NT
      f4 o;
      int gi = f4i * 4;
#pragma unroll
      for (int j = 0; j < 4; ++j) {
        float x = v[j];
        bool keep = (gi + j == i0) || (x >= T);
        float ov = keep ? x : NEG_INF;
        o[j] = ov;
        float y = ov + g[j];                         // -inf + finite = -inf
        if (y > abm[j]) { abm[j] = y; abi[j] = gi + j; }
      }
      __builtin_nontemporal_store(o, orow + f4i);    // write-once: NT
      int tn = t + DEPTH;
      async_issue_guarded(my_lds + (unsigned)((tn & (DEPTH - 1)) * 512),
                          (unsigned)((my_f4 + tn * 32) * 16), rowbase);
    }
    for (; t < TILES; ++t) {
      wait_async_le(TILES - t - 1);
      f4 v = stage[my_slot + (t & (DEPTH - 1)) * 32];
      int f4i = my_f4 + t * 32;
      f4 g = __builtin_nontemporal_load(grow + f4i);
      f4 o;
      int gi = f4i * 4;
#pragma unroll
      for (int j = 0; j < 4; ++j) {
        float x = v[j];
        bool keep = (gi + j == i0) || (x >= T);
        float ov = keep ? x : NEG_INF;
        o[j] = ov;
        float y = ov + g[j];
        if (y > abm[j]) { abm[j] = y; abi[j] = gi + j; }
      }
      __builtin_nontemporal_store(o, orow + f4i);
    }
  }
  float bm = abm[0]; int bi = abi[0];
#pragma unroll
  for (int j = 1; j < 4; ++j) {
    if (abm[j] > bm || (abm[j] == bm && abi[j] < bi)) { bm = abm[j]; bi = abi[j]; }
  }
#pragma unroll
  for (int off = 16; off >= 1; off >>= 1) {
    float yo = __shfl_xor(bm, off, 32);
    int   io = __shfl_xor(bi, off, 32);
    if (yo > bm || (yo == bm && io < bi)) { bm = yo; bi = io; }
  }
  if (lane == 0) { red_f0[w] = bm; red_i[w] = bi; }
  __syncthreads();
  if (tid == 0) {
    float BM = red_f0[0]; int BI = red_i[0];
    for (int i = 1; i < WAVES; ++i) {
      if (red_f0[i] > BM || (red_f0[i] == BM && red_i[i] < BI)) {
        BM = red_f0[i]; BI = red_i[i];
      }
    }
    out_tokens[row] = (float)BI;
  }
}

extern "C" void kernel_launch(void* const* d_in, const int* in_sizes, int n_in,
                              void* d_out, int out_size, void* d_ws, size_t ws_size,
                              hipStream_t stream) {
  const float* logits = (const float*)d_in[0];
  const float* eps    = (const float*)d_in[1];
  const float* eta    = (const float*)d_in[2];
  const float* minp   = (const float*)d_in[3];
  const float* gum    = (const float*)d_in[4];
  float* out_logits = (float*)d_out;
  float* out_tokens = out_logits + (size_t)B * V;   // tuple order: logits, tokens
  Sampler_50706383897220_kernel<<<B, THREADS, 0, stream>>>(
      logits, eps, eta, minp, gum, out_logits, out_tokens);
}